// BEVFeatureExtractor_50199577755857
// MI455X (gfx1250) — compile-verified
//
#include <hip/hip_runtime.h>
#include <hip/hip_bf16.h>
#include <math.h>

typedef __attribute__((ext_vector_type(2))) float v2f;
typedef __attribute__((ext_vector_type(8))) float v8f;

// Problem constants (fixed by the reference setup)
#define BN     4
#define HH     180
#define WW     180
#define CC     256
#define NPTS   10240      // points per batch
#define SEC    2048       // NPTS / NUM_POINT
#define PCOLS  1280       // NUM_POINT * CC
#define GROUPS_PER_B 640  // NPTS / 16
#define SPLITS 4          // channel-tile splits per 16-point group

static __device__ __forceinline__ int clampi(int v, int lo, int hi) {
    return v < lo ? lo : (v > hi ? hi : v);
}

__global__ __launch_bounds__(256)
void bev_extract_wmma(const float* __restrict__ bev,
                      const float* __restrict__ centers,
                      float* __restrict__ out) {
    const int tid  = threadIdx.x;
    const int l    = tid & 31;          // lane in wave32
    const int m    = l & 15;           // point-in-group / channel-in-tile id
    const int hi16 = (l >> 4) & 1;     // lane half selects K=0,1 vs K=2,3

    const int wid   = blockIdx.x * 8 + (tid >> 5);   // global wave job id
    const int group = wid >> 2;                      // 16-point group
    const int q     = wid & (SPLITS - 1);            // which 4 channel tiles
    const int b     = group / GROUPS_PER_B;
    const int rem   = group - b * GROUPS_PER_B;
    const int n0    = rem << 4;                      // first point of group

    // ---- per-lane: coords + bilinear weights of point (n0 + m) ----
    const float* cp = centers + ((size_t)b * NPTS + n0 + m) * 2;
    const float cx = cp[0];
    const float cy = cp[1];
    // match reference numerics: (c - (-54)) / 0.075 / 8
    const float xs = (cx - (-54.0f)) / 0.075f / 8.0f;
    const float ys = (cy - (-54.0f)) / 0.075f / 8.0f;
    int x0 = (int)floorf(xs); int x1 = x0 + 1;
    int y0 = (int)floorf(ys); int y1 = y0 + 1;
    x0 = clampi(x0, 0, WW - 1); x1 = clampi(x1, 0, WW - 1);
    y0 = clampi(y0, 0, HH - 1); y1 = clampi(y1, 0, HH - 1);
    const float x0f = (float)x0, x1f = (float)x1;
    const float y0f = (float)y0, y1f = (float)y1;
    const float wa = (x1f - xs) * (y1f - ys);   // Ia = im[y0,x0]
    const float wb = (x1f - xs) * (ys - y0f);   // Ib = im[y1,x0]
    const float wc = (xs - x0f) * (y1f - ys);   // Ic = im[y0,x1]
    const float wd = (xs - x0f) * (ys - y0f);   // Id = im[y1,x1]

    const int offA = (y0 * WW + x0) << 8;   // *256 channels
    const int offB = (y1 * WW + x0) << 8;
    const int offC = (y0 * WW + x1) << 8;
    const int offD = (y1 * WW + x1) << 8;

    // A-matrix lane roles: lanes 0-15 carry K=0,1 (wa,wb); 16-31 carry K=2,3 (wc,wd)
    const int   selOff0 = hi16 ? offC : offA;
    const int   selOff1 = hi16 ? offD : offB;
    const float selW0   = hi16 ? wc : wa;
    const float selW1   = hi16 ? wd : wb;

    const float* imb = bev + (size_t)b * (HH * WW * CC);

    // output addressing: point n -> row b*SEC + (n % SEC), col (n / SEC)*CC + c
    const int p  = n0 >> 11;       // n0 / SEC  (group never straddles a slab)
    const int jr = n0 & (SEC - 1);
    const size_t outBase =
        (size_t)(b * SEC + jr + hi16 * 8) * PCOLS + (size_t)(p * CC + m);

    const int srcHalf = l & 16;

    for (int ti = 0; ti < 4; ++ti) {
        const int t16 = (q * 4 + ti) * 16;   // channel tile base
        v8f acc = {0.f, 0.f, 0.f, 0.f, 0.f, 0.f, 0.f, 0.f};

        // 16 chained WMMAs: WMMA j contributes row j = point n0+j
#pragma unroll
        for (int j = 0; j < 16; ++j) {
            const int o0 = __shfl(selOff0, srcHalf | j, 32);
            const int o1 = __shfl(selOff1, srcHalf | j, 32);
            const float a0 = (m == j) ? selW0 : 0.0f;
            const float a1 = (m == j) ? selW1 : 0.0f;
            const float b0 = imb[o0 + t16 + m];   // B rows K=0 / K=2
            const float b1 = imb[o1 + t16 + m];   // B rows K=1 / K=3
            v2f A;  A.x = a0; A.y = a1;
            v2f Bv; Bv.x = b0; Bv.y = b1;
            acc = __builtin_amdgcn_wmma_f32_16x16x4_f32(
                false, A, false, Bv, (short)0, acc, false, false);
        }

        // D layout: VGPR r, lanes 0-15 -> M=r ; lanes 16-31 -> M=8+r ; N=lane%16
#pragma unroll
        for (int r = 0; r < 8; ++r) {
            out[outBase + (size_t)r * PCOLS + t16] = acc[r];
        }
    }
}

extern "C" void kernel_launch(void* const* d_in, const int* in_sizes, int n_in,
                              void* d_out, int out_size, void* d_ws, size_t ws_size,
                              hipStream_t stream) {
    (void)in_sizes; (void)n_in; (void)out_size; (void)d_ws; (void)ws_size;
    const float* bev     = (const float*)d_in[0];   // (4,180,180,256) f32
    const float* centers = (const float*)d_in[1];   // (4,10240,2)    f32
    float* out           = (float*)d_out;           // (4,2048,1280)  f32

    // 40960 points / 16 per group * 4 tile-splits = 10240 wave jobs; 8 waves/block
    bev_extract_wmma<<<1280, 256, 0, stream>>>(bev, centers, out);
}